// GridFeatureToPoint_49435073577160
// MI455X (gfx1250) — compile-verified
//
#include <hip/hip_runtime.h>
#include <math.h>

typedef float v2f __attribute__((ext_vector_type(2)));
typedef float v4f __attribute__((ext_vector_type(4)));
typedef float v8f __attribute__((ext_vector_type(8)));

namespace {
constexpr int CG = 64;              // grid channels
constexpr int CP = 32;              // point-feat channels
constexpr int CO = 64;              // output channels
constexpr int KD = CP + CG;         // 96 = GEMM K dim
constexpr int RR = 96;              // grid resolution
constexpr int CS = RR * RR * RR;    // channel stride = 884736
constexpr int WAVES = 6;
constexpr int BLOCK = WAVES * 32;   // 192 threads
constexpr int FSTR  = 100;          // padded feats row stride (floats), conflict-free
constexpr int HSTR  = 68;          // padded h row stride for LN staging
constexpr float LN_EPS = 1e-5f;
}

__global__ __launch_bounds__(BLOCK)
void grid_feat_point_kernel(const float* __restrict__ grid,
                            const float* __restrict__ vertices,
                            const float* __restrict__ point_feats,
                            const float* __restrict__ W_lin,
                            const float* __restrict__ b_lin,
                            const float* __restrict__ gamma,
                            const float* __restrict__ beta,
                            float* __restrict__ out,
                            int N, int ntiles)
{
    __shared__ float sW[KD * CO];           // Wt[k][n], 24576 B
    __shared__ float sGB[2 * CO];           // gamma | beta
    __shared__ float sF[WAVES * 16 * FSTR]; // per-wave 16x96 feature tiles (padded)

    const int tid  = threadIdx.x;
    const int wv   = tid >> 5;
    const int lane = tid & 31;
    const int p16  = lane & 15;             // point-in-tile / N-column
    const int g    = lane >> 4;             // half-wave id

    // Stage W transposed (Wt[k][n] = W_lin[n][k]) and gamma/beta into LDS once.
    for (int i = tid; i < KD * CO; i += BLOCK) {
        int k = i / CO, n = i - k * CO;
        sW[i] = W_lin[n * KD + k];
    }
    for (int i = tid; i < 2 * CO; i += BLOCK)
        sGB[i] = (i < CO) ? gamma[i] : beta[i - CO];
    __syncthreads();

    float* sFw = sF + wv * (16 * FSTR);

    // Bias values for this lane's output column (N = p16) in the 4 sub-tiles.
    const float bb0 = b_lin[p16];
    const float bb1 = b_lin[16 + p16];
    const float bb2 = b_lin[32 + p16];
    const float bb3 = b_lin[48 + p16];

    for (int tile = blockIdx.x * WAVES + wv; tile < ntiles; tile += gridDim.x * WAVES) {
        const int pidx = tile * 16 + p16;
        const int pc   = pidx < N ? pidx : N - 1;

        //----- phase A: stage point_feats[pc][g*16 .. g*16+15] into feats row
        //      (streamed once -> non-temporal so it never evicts grid lines from L2)
        {
            const v4f* pf = (const v4f*)(point_feats + (size_t)pc * CP + g * 16);
            v4f* dst = (v4f*)&sFw[p16 * FSTR + g * 16];
            #pragma unroll
            for (int q = 0; q < 4; ++q) dst[q] = __builtin_nontemporal_load(pf + q);
        }

        //----- phase B: trilinear gather of 32 channels per lane
        {
            const float vx = __builtin_nontemporal_load(vertices + (size_t)pc * 3 + 0);
            const float vy = __builtin_nontemporal_load(vertices + (size_t)pc * 3 + 1);
            const float vz = __builtin_nontemporal_load(vertices + (size_t)pc * 3 + 2);
            // reference: f = ((v*2-1)+1)*0.5*(R-1) = v*95
            const float fx = vx * 95.0f, fy = vy * 95.0f, fz = vz * 95.0f;
            const float fx0 = floorf(fx), fy0 = floorf(fy), fz0 = floorf(fz);
            int ix0 = (int)fx0, iy0 = (int)fy0, iz0 = (int)fz0;
            ix0 = ix0 < 0 ? 0 : (ix0 > RR - 1 ? RR - 1 : ix0);
            iy0 = iy0 < 0 ? 0 : (iy0 > RR - 1 ? RR - 1 : iy0);
            iz0 = iz0 < 0 ? 0 : (iz0 > RR - 1 ? RR - 1 : iz0);
            const int ix1 = ix0 + 1 > RR - 1 ? RR - 1 : ix0 + 1;
            const int iy1 = iy0 + 1 > RR - 1 ? RR - 1 : iy0 + 1;
            const int iz1 = iz0 + 1 > RR - 1 ? RR - 1 : iz0 + 1;
            const float wx1 = fx - fx0, wy1 = fy - fy0, wz1 = fz - fz0;
            const float wx0 = 1.0f - wx1, wy0 = 1.0f - wy1, wz0 = 1.0f - wz1;

            const int b00 = (iz0 * RR + iy0) * RR;
            const int b01 = (iz0 * RR + iy1) * RR;
            const int b10 = (iz1 * RR + iy0) * RR;
            const int b11 = (iz1 * RR + iy1) * RR;
            const int o0 = b00 + ix0, o1 = b00 + ix1;
            const int o2 = b01 + ix0, o3 = b01 + ix1;
            const int o4 = b10 + ix0, o5 = b10 + ix1;
            const int o6 = b11 + ix0, o7 = b11 + ix1;
            const float w0 = wx0 * wy0 * wz0, w1 = wx1 * wy0 * wz0;
            const float w2 = wx0 * wy1 * wz0, w3 = wx1 * wy1 * wz0;
            const float w4 = wx0 * wy0 * wz1, w5 = wx1 * wy0 * wz1;
            const float w6 = wx0 * wy1 * wz1, w7 = wx1 * wy1 * wz1;

            // grid loads stay regular-temporal: the 192 MB L2 holds most of the
            // 226 MB grid, and L2 reuse is the only locality this gather has.
            const float* gp = grid + (size_t)(g * 32) * CS;
            float* frow = &sFw[p16 * FSTR + CP + g * 32];
            for (int cc = 0; cc < 32; ++cc) {
                float s = gp[o0] * w0 + gp[o1] * w1 + gp[o2] * w2 + gp[o3] * w3
                        + gp[o4] * w4 + gp[o5] * w5 + gp[o6] * w6 + gp[o7] * w7;
                frow[cc] = s;
                gp += CS;
            }
        }

        asm volatile("s_wait_dscnt 0x0" ::: "memory");

        //----- phase C: [16 x 96] x [96 x 64] GEMM via V_WMMA_F32_16X16X4_F32
        v8f acc0 = {}, acc1 = {}, acc2 = {}, acc3 = {};
        {
            const float* Arow = &sFw[p16 * FSTR + 2 * g];
            #pragma unroll 4
            for (int k0 = 0; k0 < KD; k0 += 4) {
                v2f a; a.x = Arow[k0]; a.y = Arow[k0 + 1];
                const float* Bk = &sW[(k0 + 2 * g) * CO + p16];
                v2f b0, b1, b2, b3;
                b0.x = Bk[0];   b0.y = Bk[CO];
                b1.x = Bk[16];  b1.y = Bk[CO + 16];
                b2.x = Bk[32];  b2.y = Bk[CO + 32];
                b3.x = Bk[48];  b3.y = Bk[CO + 48];
                acc0 = __builtin_amdgcn_wmma_f32_16x16x4_f32(false, a, false, b0, (short)0, acc0, false, false);
                acc1 = __builtin_amdgcn_wmma_f32_16x16x4_f32(false, a, false, b1, (short)0, acc1, false, false);
                acc2 = __builtin_amdgcn_wmma_f32_16x16x4_f32(false, a, false, b2, (short)0, acc2, false, false);
                acc3 = __builtin_amdgcn_wmma_f32_16x16x4_f32(false, a, false, b3, (short)0, acc3, false, false);
            }
        }

        //----- phase D: stage h (+bias) to LDS (reuse this wave's slice)
        float* sH = sFw;
        #pragma unroll
        for (int r = 0; r < 8; ++r) {
            const int m = r + 8 * g;           // D layout: lanes 0-15 -> M=r, 16-31 -> M=r+8
            sH[m * HSTR +      p16] = acc0[r] + bb0;
            sH[m * HSTR + 16 + p16] = acc1[r] + bb1;
            sH[m * HSTR + 32 + p16] = acc2[r] + bb2;
            sH[m * HSTR + 48 + p16] = acc3[r] + bb3;
        }
        asm volatile("s_wait_dscnt 0x0" ::: "memory");

        //----- phase E: LayerNorm over 64 channels of point p16
        float sum = 0.f, sumsq = 0.f;
        const float* hrow = &sH[p16 * HSTR];
        #pragma unroll
        for (int q = 0; q < 16; ++q) {
            v4f h4 = *(const v4f*)&hrow[q * 4];
            sum   += h4.x + h4.y + h4.z + h4.w;
            sumsq += h4.x * h4.x + h4.y * h4.y + h4.z * h4.z + h4.w * h4.w;
        }
        const float mu = sum * (1.0f / CO);
        float var = sumsq * (1.0f / CO) - mu * mu;
        var = var < 0.f ? 0.f : var;
        const float rstd = rsqrtf(var + LN_EPS);

        if (pidx < N) {
            // output is write-once, never re-read -> non-temporal stores keep L2 for the grid
            v4f* op = (v4f*)(out + (size_t)pidx * CO + g * 32);
            const float* hseg = hrow + g * 32;
            const float* gseg = &sGB[g * 32];
            const float* bseg = &sGB[CO + g * 32];
            #pragma unroll
            for (int q = 0; q < 8; ++q) {
                v4f h4 = *(const v4f*)&hseg[q * 4];
                v4f g4 = *(const v4f*)&gseg[q * 4];
                v4f b4 = *(const v4f*)&bseg[q * 4];
                v4f o;
                o.x = (h4.x - mu) * rstd * g4.x + b4.x;
                o.y = (h4.y - mu) * rstd * g4.y + b4.y;
                o.z = (h4.z - mu) * rstd * g4.z + b4.z;
                o.w = (h4.w - mu) * rstd * g4.w + b4.w;
                __builtin_nontemporal_store(o, op + q);
            }
        }
        asm volatile("s_wait_dscnt 0x0" ::: "memory");  // before next tile reuses sF
    }
}

extern "C" void kernel_launch(void* const* d_in, const int* in_sizes, int n_in,
                              void* d_out, int out_size, void* d_ws, size_t ws_size,
                              hipStream_t stream) {
    (void)n_in; (void)out_size; (void)d_ws; (void)ws_size;
    const float* grid        = (const float*)d_in[0];
    const float* vertices    = (const float*)d_in[1];
    const float* point_feats = (const float*)d_in[2];
    const float* W_lin       = (const float*)d_in[3];
    const float* b_lin       = (const float*)d_in[4];
    const float* gamma       = (const float*)d_in[5];
    const float* beta        = (const float*)d_in[6];
    float* out = (float*)d_out;

    const int N = in_sizes[1] / 3;
    const int ntiles = (N + 15) / 16;
    int blocks = (ntiles + WAVES - 1) / WAVES;
    if (blocks > 2048) blocks = 2048;

    grid_feat_point_kernel<<<blocks, BLOCK, 0, stream>>>(
        grid, vertices, point_feats, W_lin, b_lin, gamma, beta, out, N, ntiles);
}